// GraphNodeEdgeConvolution_72301479461280
// MI455X (gfx1250) — compile-verified
//
#include <hip/hip_runtime.h>

#define NN 2048
#define KE 8
#define FNF 16
#define OO 32

typedef __attribute__((ext_vector_type(2))) float v2f;
typedef __attribute__((ext_vector_type(4))) float v4f;
typedef __attribute__((ext_vector_type(8))) float v8f;

// workspace layout (floats): r_inv [0,2048) | col_sum [2048,4096) | agg_e [4096, 4096+2048*8)

__global__ void zero_ws_kernel(float* __restrict__ ws) {
    int t = blockIdx.x * blockDim.x + threadIdx.x;
    const int n = NN + NN * KE;            // col_sum + agg_e
    if (t < n) ws[NN + t] = 0.0f;
}

// one wave32 per row: r_inv[i] = 1/sum_j adj[i,j] (0 if non-finite); b128 loads
__global__ void row_inv_kernel(const float* __restrict__ adj, float* __restrict__ r_inv) {
    int wave = (blockIdx.x * blockDim.x + threadIdx.x) >> 5;
    int lane = threadIdx.x & 31;
    const v4f* row = (const v4f*)(adj + (size_t)wave * NN);   // 512 float4 per row
    float s = 0.0f;
    #pragma unroll 4
    for (int m = lane; m < NN / 4; m += 32) {
        v4f v = row[m];
        s += v.x + v.y + v.z + v.w;
    }
    #pragma unroll
    for (int off = 16; off >= 1; off >>= 1) s += __shfl_xor(s, off, 32);
    if (lane == 0) {
        float inv = 1.0f / s;
        if (!isfinite(inv)) inv = 0.0f;
        r_inv[wave] = inv;
    }
}

// main 134MB stream: j-tile = 128, thread = (jj, k-quad); b128 non-temporal ef loads
__global__ void agg_kernel(const float* __restrict__ adj,
                           const float* __restrict__ ef,
                           const float* __restrict__ r_inv,
                           float* __restrict__ col_sum,
                           float* __restrict__ agg_e) {
    const int jj = threadIdx.x >> 1;          // 0..127
    const int kq = (threadIdx.x & 1) << 2;    // 0 or 4
    const int j  = blockIdx.x * 128 + jj;
    const int i0 = blockIdx.y * 64;
    v4f acc = {0.0f, 0.0f, 0.0f, 0.0f};
    float csum = 0.0f;
    #pragma unroll 4
    for (int i = i0; i < i0 + 64; ++i) {
        float w = r_inv[i] * adj[(size_t)i * NN + j];
        // single-use stream: non-temporal keeps adj/r_inv resident in the 192MB L2
        const v4f* p = (const v4f*)(ef + ((size_t)i * NN + j) * KE + kq);
        v4f e = __builtin_nontemporal_load(p);
        acc.x += w * e.x;
        acc.y += w * e.y;
        acc.z += w * e.z;
        acc.w += w * e.w;
        csum += w;
    }
    float* dst = &agg_e[j * KE + kq];
    atomicAdd(dst + 0, acc.x);
    atomicAdd(dst + 1, acc.y);
    atomicAdd(dst + 2, acc.z);
    atomicAdd(dst + 3, acc.w);
    if (kq == 0) atomicAdd(&col_sum[j], csum);
}

// out = [agg_e | col_sum*nf] (2048x24) @ weight (24x32), one 16x16 tile per wave
// via 6 chained V_WMMA_F32_16X16X4_F32. Branchless: K phases split explicitly.
__global__ void finalize_wmma_kernel(const float* __restrict__ nf,
                                     const float* __restrict__ weight,
                                     const float* __restrict__ col_sum,
                                     const float* __restrict__ agg_e,
                                     float* __restrict__ out) {
    const int lane = threadIdx.x;     // blockDim.x == 32, EXEC all ones
    const int m    = lane & 15;
    const int half = lane >> 4;       // selects K{0,1} vs K{2,3} within a 4-chunk
    const int wid  = blockIdx.x;      // 0..255
    const int j0   = (wid >> 1) * 16;
    const int o0   = (wid & 1) * 16;
    const int jA   = j0 + m;          // A-matrix row for this lane
    const float cs = col_sum[jA];

    v8f c = {};
    // phase A: K = 0..7  -> X columns from agg_e
    #pragma unroll
    for (int kk = 0; kk < 2; ++kk) {
        const int ka = kk * 4 + half * 2;
        v2f a, b;
        a.x = agg_e[jA * KE + ka];
        a.y = agg_e[jA * KE + ka + 1];
        b.x = weight[ka * OO + o0 + m];
        b.y = weight[(ka + 1) * OO + o0 + m];
        c = __builtin_amdgcn_wmma_f32_16x16x4_f32(
                false, a, false, b, (short)0, c, false, false);
    }
    // phase B: K = 8..23 -> X columns = col_sum * node_features
    #pragma unroll
    for (int kk = 0; kk < 4; ++kk) {
        const int kb = kk * 4 + half * 2;       // 0..15 into nf columns
        v2f a, b;
        a.x = cs * nf[jA * FNF + kb];
        a.y = cs * nf[jA * FNF + kb + 1];
        b.x = weight[(kb + KE) * OO + o0 + m];
        b.y = weight[(kb + KE + 1) * OO + o0 + m];
        c = __builtin_amdgcn_wmma_f32_16x16x4_f32(
                false, a, false, b, (short)0, c, false, false);
    }
    // 32-bit C/D 16x16 layout: VGPR v, lanes 0-15 -> M=v, lanes 16-31 -> M=v+8
    #pragma unroll
    for (int v = 0; v < 8; ++v) {
        int row = j0 + v + half * 8;
        out[row * OO + o0 + m] = c[v];
    }
}

extern "C" void kernel_launch(void* const* d_in, const int* in_sizes, int n_in,
                              void* d_out, int out_size, void* d_ws, size_t ws_size,
                              hipStream_t stream) {
    const float* nf     = (const float*)d_in[0];  // (2048, 16)
    const float* ef     = (const float*)d_in[1];  // (2048, 2048, 8)
    const float* adj    = (const float*)d_in[2];  // (2048, 2048)
    const float* weight = (const float*)d_in[3];  // (24, 32)
    float* out = (float*)d_out;                    // (2048, 32)
    float* ws  = (float*)d_ws;

    float* r_inv   = ws;
    float* col_sum = ws + NN;
    float* agg_e   = ws + 2 * NN;

    // 1) zero accumulators (col_sum + agg_e = 18432 floats)
    zero_ws_kernel<<<(NN + NN * KE + 255) / 256, 256, 0, stream>>>(ws);
    // 2) row sums -> r_inv : 2048 waves, 8 waves/block
    row_inv_kernel<<<NN / 8, 256, 0, stream>>>(adj, r_inv);
    // 3) main stream: 16 j-tiles (128 wide) x 32 i-chunks (64 rows)
    agg_kernel<<<dim3(16, 32), 256, 0, stream>>>(adj, ef, r_inv, col_sum, agg_e);
    // 4) WMMA finalize: 256 tiles of 16x16, one wave each
    finalize_wmma_kernel<<<256, 32, 0, stream>>>(nf, weight, col_sum, agg_e, out);
}